// LIFSpike_3745211482710
// MI455X (gfx1250) — compile-verified
//
#include <hip/hip_runtime.h>
#include <stdint.h>
#include <utility>

// LIF scan: u_t = TAU*u_{t-1}*(1-o_{t-1}) + x_t ; o_t = (u_t > VTH)
#define LIF_TAU 0.3f
#define LIF_VTH 0.3f
#define LIF_T   50          // time steps, innermost contiguous dim
#define TPB     128         // 4 waves per block (wave32)
#define ITERS   4           // sequences per thread (double-buffer ping-pong)
#define CH      25          // 25 x 8B chunks = 200 B = 50 floats per sequence

typedef float v2f __attribute__((ext_vector_type(2)));
typedef int   v2i __attribute__((ext_vector_type(2)));

#if __has_builtin(__builtin_amdgcn_global_load_async_to_lds_b64)
#define HAVE_ASYNC_LDS 1
#else
#define HAVE_ASYNC_LDS 0
#endif

#if __has_builtin(__builtin_amdgcn_s_wait_asynccnt)
#define WAIT_ASYNC(n) __builtin_amdgcn_s_wait_asynccnt(n)
#else
#define WAIT_ASYNC(n) asm volatile("s_wait_asynccnt %0" ::"n"(n) : "memory")
#endif

#define AS_GLOBAL __attribute__((address_space(1)))
#define AS_LOCAL  __attribute__((address_space(3)))

// Builtin signature is (global v2i*, local v2i*, imm offset, imm cpol).
typedef AS_GLOBAL v2i* g_v2i_ptr;  // pointer to v2i in addrspace(1)
typedef AS_LOCAL  v2i* l_v2i_ptr;  // pointer to v2i in addrspace(3)

#if HAVE_ASYNC_LDS
// Issue CH async 8-byte copies sharing ONE global address pair and ONE LDS
// address VGPR: the per-chunk byte offset is the instruction's immediate
// IOFFSET, which hardware adds to BOTH the LDS and global addresses
// (same offset on both sides). Template params are ICEs, so the
// frontend accepts them as the required constant offset argument.
template <int... Ks>
__device__ __forceinline__ void async_copy_200B(g_v2i_ptr gp, l_v2i_ptr lp,
                                                std::integer_sequence<int, Ks...>) {
  (__builtin_amdgcn_global_load_async_to_lds_b64(gp, lp, Ks * 8, 0), ...);
}
#endif

// One LIF step, numerically identical to the reference:
// reference: t = (TAU*u) rounded; t *= (1-o) which is exact *1 or *0; u = t + x.
// here: dec = o ? 0 : TAU (exact), u = dec*u + x with contraction OFF.
__device__ __forceinline__ void lif_step(float x_t, float& u, float& dec, float& o) {
#pragma clang fp contract(off)
  float m = dec * u;
  u = m + x_t;
  bool sp = (u > LIF_VTH);
  o   = sp ? 1.0f : 0.0f;
  dec = sp ? 0.0f : LIF_TAU;
}

__global__ __launch_bounds__(TPB) void lif_spike_kernel(const float* __restrict__ x,
                                                        float* __restrict__ out,
                                                        int nseq) {
#if HAVE_ASYNC_LDS
  // Double-buffered async staging: 2 * 128 * 200B = 51,200 B LDS per block.
  __shared__ float lds[2][TPB * LIF_T];
  const int tid  = threadIdx.x;
  const int base = blockIdx.x * (TPB * ITERS);

  auto issue_tile = [&](int it, int buf) {
    const int s = base + it * TPB + tid;
    if (s < nseq) {
      g_v2i_ptr gp = (g_v2i_ptr)(uintptr_t)(x + (size_t)s * LIF_T);
      l_v2i_ptr lp = (l_v2i_ptr)(uintptr_t)(&lds[buf][tid * LIF_T]);
      async_copy_200B(gp, lp, std::make_integer_sequence<int, CH>{});
    }
  };

  issue_tile(0, 0);
#pragma unroll
  for (int it = 0; it < ITERS; ++it) {
    const bool has_next = (it + 1 < ITERS) && (base + (it + 1) * TPB < nseq);
    if (has_next) issue_tile(it + 1, (it + 1) & 1);
    // Async loads complete in order: <=CH outstanding means current tile landed.
    if (has_next) { WAIT_ASYNC(CH); } else { WAIT_ASYNC(0); }
    asm volatile("" ::: "memory");  // compiler fence: LDS now holds new data

    const int s = base + it * TPB + tid;
    if (s < nseq) {
      // Lane stride = 50 dwords -> ds_load_b64 is bank-conflict free
      // (gcd(50,64)=2; 50*d mod 64 never hits {0,+-1} for d=1..31).
      const AS_LOCAL v2f* lp =
          (const AS_LOCAL v2f*)(uintptr_t)(&lds[it & 1][tid * LIF_T]);
      float* o = out + (size_t)s * LIF_T;
      float u = 0.0f, dec = LIF_TAU;
#pragma unroll
      for (int k = 0; k < CH; ++k) {
        v2f xv = lp[k];
        float o0, o1;
        lif_step(xv.x, u, dec, o0);
        lif_step(xv.y, u, dec, o1);
        v2f ov;
        ov.x = o0;
        ov.y = o1;
        *(v2f*)(o + 2 * k) = ov;  // 8B-aligned streaming store (200B stride)
      }
    }
  }
#else
  // Fallback: direct global->register streaming with gfx1250 prefetch.
  const int tid  = threadIdx.x;
  const int base = blockIdx.x * (TPB * ITERS);
  for (int it = 0; it < ITERS; ++it) {
    const int s = base + it * TPB + tid;
    if (s >= nseq) continue;
    const float* g = x + (size_t)s * LIF_T;
    if (it + 1 < ITERS && (s + TPB) < nseq)
      __builtin_prefetch(g + (size_t)TPB * LIF_T, 0, 0);  // global_prefetch_b8
    float* o = out + (size_t)s * LIF_T;
    float u = 0.0f, dec = LIF_TAU;
#pragma unroll
    for (int k = 0; k < CH; ++k) {
      v2f xv = *(const v2f*)(g + 2 * k);
      float o0, o1;
      lif_step(xv.x, u, dec, o0);
      lif_step(xv.y, u, dec, o1);
      v2f ov;
      ov.x = o0;
      ov.y = o1;
      *(v2f*)(o + 2 * k) = ov;
    }
  }
#endif
}

extern "C" void kernel_launch(void* const* d_in, const int* in_sizes, int n_in,
                              void* d_out, int out_size, void* d_ws, size_t ws_size,
                              hipStream_t stream) {
  const float* x = (const float*)d_in[0];  // [16,64,32,32,50] fp32
  // d_in[1] = ksi, unused by the reference computation.
  float* out = (float*)d_out;

  const int total = in_sizes[0];        // 52,428,800 elements
  const int nseq  = total / LIF_T;      // 1,048,576 independent sequences
  int blocks = (nseq + TPB * ITERS - 1) / (TPB * ITERS);
  if (blocks < 1) blocks = 1;

  lif_spike_kernel<<<blocks, TPB, 0, stream>>>(x, out, nseq);
}